// VectorQuantizer_16303695856141
// MI455X (gfx1250) — compile-verified
//
#include <hip/hip_runtime.h>

typedef __attribute__((ext_vector_type(2))) float v2f;
typedef __attribute__((ext_vector_type(4))) float v4f;
typedef __attribute__((ext_vector_type(8))) float v8f;
typedef __attribute__((ext_vector_type(4))) int   v4i;

#define N_ROWS      65536
#define CDIM        64
#define K_CODES     1024
#define TILE_CODES  64                          // codes per LDS tile (16 KB)
#define N_TILES     (K_CODES / TILE_CODES)      // 16
#define SUBT        16                          // codes per WMMA tile
#define WAVES       8
#define ROWS_PER_WAVE 16
#define ROWS_PER_BLOCK (WAVES * ROWS_PER_WAVE)  // 128
#define N_ELEMS     (N_ROWS * CDIM)             // 4194304
#define TILE_FLOATS (TILE_CODES * CDIM)         // 4096
#define LOADS_PER_THREAD (TILE_FLOATS / 4 / 256) // 4 x b128 per thread per tile

#if __has_builtin(__builtin_amdgcn_global_load_async_to_lds_b128) && \
    __has_builtin(__builtin_amdgcn_s_wait_asynccnt)
#define VQ_ASYNC 1
typedef __attribute__((address_space(1))) v4i* vq_gp;   // global int4*
typedef __attribute__((address_space(3))) v4i* vq_lp;   // LDS int4*

__device__ __forceinline__ vq_gp vq_to_global(const float* p) {
    // global addresses are bit-identical in generic and AS(1)
    return (vq_gp)(uintptr_t)p;
}
__device__ __forceinline__ vq_lp vq_to_lds(float* p) {
    // generic LDS pointer low 32 bits == LDS offset (ISA aperture rule)
    return (vq_lp)(uint32_t)(uintptr_t)p;
}
#else
#define VQ_ASYNC 0
#endif

// ---------------------------------------------------------------------------
// Prep: per-code squared norms -> ws[0..1023]; zero loss accumulator ws[1024]
// ---------------------------------------------------------------------------
__global__ __launch_bounds__(256) void vq_prep(const float* __restrict__ emb,
                                               float* __restrict__ ws) {
    int t = blockIdx.x * blockDim.x + threadIdx.x;   // 1024 threads total
    if (t < K_CODES) {
        const v4f* e4 = (const v4f*)(emb + (size_t)t * CDIM);
        float s = 0.0f;
#pragma unroll
        for (int c = 0; c < CDIM / 4; ++c) {
            v4f v = e4[c];
            s += v.x * v.x + v.y * v.y + v.z * v.z + v.w * v.w;
        }
        ws[t] = s;
    }
    if (t == 0) ws[K_CODES] = 0.0f;
}

// Issue one codebook tile (64 codes x 64 ch = 16 KB) into an LDS buffer.
__device__ __forceinline__ void vq_issue_tile(const float* __restrict__ emb,
                                              float* __restrict__ lds_buf,
                                              int tile, int tid) {
    const float* src = emb + (size_t)tile * TILE_FLOATS + tid * 4;
    float*       dst = lds_buf + tid * 4;
#if VQ_ASYNC
#pragma unroll
    for (int i = 0; i < LOADS_PER_THREAD; ++i)
        __builtin_amdgcn_global_load_async_to_lds_b128(
            vq_to_global(src + i * 1024), vq_to_lds(dst + i * 1024), 0, 0);
#else
#pragma unroll
    for (int i = 0; i < LOADS_PER_THREAD; ++i)
        *(v4f*)(dst + i * 1024) = *(const v4f*)(src + i * 1024);
#endif
}

// ---------------------------------------------------------------------------
// Main fused kernel: distances via V_WMMA_F32_16X16X4_F32 with double-buffered
// async LDS codebook tiles; argmin; gather; quantized write; loss partials.
// ---------------------------------------------------------------------------
__global__ __launch_bounds__(256) void vq_main(const float* __restrict__ x,
                                               const float* __restrict__ emb,
                                               const float* __restrict__ ws_norm,
                                               float* __restrict__ out,
                                               float* __restrict__ loss_accum) {
    __shared__ float s_emb[2][TILE_FLOATS];      // 2 x 16 KB codebook tiles
    __shared__ float s_norm[K_CODES];            // 4 KB  ||e||^2

    const int tid  = threadIdx.x;
    const int lane = tid & 31;
    const int wave = tid >> 5;
    const int half = lane >> 4;      // 0: lanes 0-15, 1: lanes 16-31
    const int l15  = lane & 15;
    const int r0   = blockIdx.x * ROWS_PER_BLOCK + wave * ROWS_PER_WAVE;

    // Prefetch tile 0 into buffer 0 (async path overlaps with A preload).
    vq_issue_tile(emb, s_emb[0], 0, tid);

    // All code norms resident in LDS for the whole kernel.
#pragma unroll
    for (int i = 0; i < K_CODES / 256; ++i)
        s_norm[tid + 256 * i] = ws_norm[tid + 256 * i];

    // Preload A fragments (16 rows x 64 channels of x) into registers.
    // ISA 32-bit A 16x4 layout: lane L holds M = L&15, VGPR v holds
    // K = v + 2*(L>>4)  -> step k needs channels 4k+2*half, 4k+2*half+1.
    const v2f* xrow2 = (const v2f*)(x + (size_t)(r0 + l15) * CDIM);
    v2f a[16];
#pragma unroll
    for (int k = 0; k < 16; ++k)
        a[k] = xrow2[2 * k + half];

    float best_s[8];
    int   best_i[8];
#pragma unroll
    for (int v = 0; v < 8; ++v) { best_s[v] = 3.4e38f; best_i[v] = 0; }

#pragma unroll 1
    for (int t = 0; t < N_TILES; ++t) {
        const int cur = t & 1;
        // Prefetch next tile into the other buffer, then wait for this tile.
        if (t + 1 < N_TILES) {
            vq_issue_tile(emb, s_emb[cur ^ 1], t + 1, tid);
#if VQ_ASYNC
            __builtin_amdgcn_s_wait_asynccnt(LOADS_PER_THREAD); // tile t landed
#endif
        } else {
#if VQ_ASYNC
            __builtin_amdgcn_s_wait_asynccnt(0);
#endif
        }
        __syncthreads();   // all waves: tile t resident, prev buffer reads done

        const float* sbuf = s_emb[cur];
#pragma unroll 1
        for (int n = 0; n < TILE_CODES / SUBT; ++n) {
            const int codeLocal = n * SUBT + l15;   // this lane's code column
            // B layout (4x16 f32): lane L holds N = L&15, VGPR v holds
            // K = v + 2*(L>>4) -> contiguous channel pair of one code row.
            const v2f* brow2 = (const v2f*)(sbuf + (size_t)codeLocal * CDIM);

            // Two independent accumulator chains to pipeline the WMMAs.
            v8f acc0 = {0.f, 0.f, 0.f, 0.f, 0.f, 0.f, 0.f, 0.f};
            v8f acc1 = {0.f, 0.f, 0.f, 0.f, 0.f, 0.f, 0.f, 0.f};
#pragma unroll
            for (int j = 0; j < 8; ++j) {
                v2f b0 = brow2[4 * j + half];
                v2f b1 = brow2[4 * j + 2 + half];
                acc0 = __builtin_amdgcn_wmma_f32_16x16x4_f32(
                    false, a[2 * j + 0], false, b0, (short)0, acc0, false, false);
                acc1 = __builtin_amdgcn_wmma_f32_16x16x4_f32(
                    false, a[2 * j + 1], false, b1, (short)0, acc1, false, false);
            }

            const int   code    = t * TILE_CODES + codeLocal;
            const float sc_base = s_norm[code];     // ||e||^2 (||x||^2 dropped)
#pragma unroll
            for (int v = 0; v < 8; ++v) {
                float sc = __builtin_fmaf(-2.0f, acc1[v],
                           __builtin_fmaf(-2.0f, acc0[v], sc_base));
                if (sc < best_s[v]) { best_s[v] = sc; best_i[v] = code; }
            }
        }
        __syncthreads();   // all waves done reading buf[cur] before re-target
    }

    // Cross-lane argmin within each 16-lane half (rows v / v+8 per C/D layout).
    // Tie-break: lower code index (matches jnp.argmin first-occurrence).
#pragma unroll
    for (int v = 0; v < 8; ++v) {
#pragma unroll
        for (int m = 1; m < 16; m <<= 1) {
            float os = __shfl_xor(best_s[v], m, 32);
            int   oi = __shfl_xor(best_i[v], m, 32);
            if (os < best_s[v] || (os == best_s[v] && oi < best_i[v])) {
                best_s[v] = os; best_i[v] = oi;
            }
        }
    }

    // Gather winning codes, write quantized output, accumulate loss partials.
    float lsum = 0.0f;
#pragma unroll
    for (int v = 0; v < 8; ++v) {
        const int idx0 = __shfl(best_i[v], 0, 32);   // row r0 + v
        const int idx1 = __shfl(best_i[v], 16, 32);  // row r0 + v + 8
        const int row0 = r0 + v;
        const int row1 = r0 + v + 8;

        v2f e0 = ((const v2f*)(emb + (size_t)idx0 * CDIM))[lane];
        v2f x0 = ((const v2f*)(x   + (size_t)row0 * CDIM))[lane];
        ((v2f*)(out + (size_t)row0 * CDIM))[lane] = e0;
        float d0x = e0.x - x0.x, d0y = e0.y - x0.y;
        lsum += d0x * d0x + d0y * d0y;

        v2f e1 = ((const v2f*)(emb + (size_t)idx1 * CDIM))[lane];
        v2f x1 = ((const v2f*)(x   + (size_t)row1 * CDIM))[lane];
        ((v2f*)(out + (size_t)row1 * CDIM))[lane] = e1;
        float d1x = e1.x - x1.x, d1y = e1.y - x1.y;
        lsum += d1x * d1x + d1y * d1y;
    }

#pragma unroll
    for (int m = 1; m < 32; m <<= 1)
        lsum += __shfl_xor(lsum, m, 32);
    if (lane == 0)
        atomicAdd(loss_accum, lsum);
}

// ---------------------------------------------------------------------------
// Finalize: loss = (1 + BETA) * mean((q - x)^2), BETA = 0.25
// ---------------------------------------------------------------------------
__global__ void vq_final(const float* __restrict__ ws, float* __restrict__ out) {
    out[N_ELEMS] = ws[K_CODES] * (1.25f / (float)N_ELEMS);
}

extern "C" void kernel_launch(void* const* d_in, const int* in_sizes, int n_in,
                              void* d_out, int out_size, void* d_ws, size_t ws_size,
                              hipStream_t stream) {
    const float* x   = (const float*)d_in[0];   // [64,32,32,64] fp32
    const float* emb = (const float*)d_in[1];   // [1024,64] fp32
    float* out = (float*)d_out;                 // quantized (4194304) + loss (1)
    float* ws  = (float*)d_ws;                  // [0..1023] norms, [1024] loss acc

    vq_prep<<<4, 256, 0, stream>>>(emb, ws);
    vq_main<<<N_ROWS / ROWS_PER_BLOCK, 256, 0, stream>>>(x, emb, ws, out, ws + K_CODES);
    vq_final<<<1, 1, 0, stream>>>(ws, out);
}